// KSVDDictionaryLearning_44530220925038
// MI455X (gfx1250) — compile-verified
//
#include <hip/hip_runtime.h>
#include <hip/hip_bf16.h>
#include <stddef.h>

typedef __attribute__((ext_vector_type(2))) float v2f;
typedef __attribute__((ext_vector_type(8))) float v8f;

#define MDIM   64
#define NDICT  512
#define BDIM   65536
#define SPAR   5
#define DIAG_EPS 1e-6f

// ---------------------------------------------------------------------------
// Kernel 0: zero the 128MB output (bandwidth-bound, float4 stores)
// ---------------------------------------------------------------------------
__global__ void zero_out_kernel(float4* __restrict__ out, int n4) {
    int i = blockIdx.x * blockDim.x + threadIdx.x;
    int stride = gridDim.x * blockDim.x;
    float4 z = make_float4(0.f, 0.f, 0.f, 0.f);
    for (; i < n4; i += stride) out[i] = z;
}

// ---------------------------------------------------------------------------
// Kernel 1: G = D^T D  (512x512, K=64) via V_WMMA_F32_16X16X4_F32
// One wave per 16x16 tile; grid (32,32), block 32.
// A-frag (16x4 f32): lane l (l&15)=M, (l>>4) selects K pair {k0,k0+1}/{k0+2,k0+3}
// ---------------------------------------------------------------------------
__global__ __launch_bounds__(32)
void gram_kernel(const float* __restrict__ D, float* __restrict__ G) {
    const int lane = threadIdx.x;
    const int lid  = lane & 15;
    const int half = lane >> 4;
    const int i0 = blockIdx.y * 16;   // row tile (dict atom index i)
    const int j0 = blockIdx.x * 16;   // col tile (dict atom index j)

    v8f acc = {};
#pragma unroll
    for (int kk = 0; kk < 16; ++kk) {
        const int kb = kk * 4 + half * 2;
        v2f a, b;
        a.x = D[(size_t)kb       * NDICT + i0 + lid];
        a.y = D[(size_t)(kb + 1) * NDICT + i0 + lid];
        b.x = D[(size_t)kb       * NDICT + j0 + lid];
        b.y = D[(size_t)(kb + 1) * NDICT + j0 + lid];
        acc = __builtin_amdgcn_wmma_f32_16x16x4_f32(
            false, a, false, b, (short)0, acc, false, false);
    }
#pragma unroll
    for (int v = 0; v < 8; ++v) {
        const int m = v + 8 * half;
        G[(size_t)(i0 + m) * NDICT + (j0 + lid)] = acc[v];
    }
}

// ---------------------------------------------------------------------------
// Kernel 2: fused dtx = D^T X (WMMA f32) + Batch-OMP per column.
// One wave32 block handles 16 batch columns.
// LDS: dtx stored as sDtx[col*513 + i] (513 padding -> conflict-free both ways)
// ---------------------------------------------------------------------------
__global__ __launch_bounds__(32)
void omp_kernel(const float* __restrict__ X, const float* __restrict__ D,
                const float* __restrict__ G, float* __restrict__ out) {
    __shared__ float sDtx[16 * 513];   // 32.8 KB

    const int lane = threadIdx.x;
    const int lid  = lane & 15;
    const int half = lane >> 4;
    const int col0 = blockIdx.x * 16;

    // ---- Phase A: dtx tile = D^T(512x64) * X(64x16cols), fp32 WMMA ----
    // Hoist the B (X) fragments for all 16 K-steps: 32 VGPRs, reused 32x.
    v2f bx[16];
#pragma unroll
    for (int kk = 0; kk < 16; ++kk) {
        const int kb = kk * 4 + half * 2;
        bx[kk].x = X[(size_t)kb       * BDIM + col0 + lid];
        bx[kk].y = X[(size_t)(kb + 1) * BDIM + col0 + lid];
    }
    for (int r = 0; r < 32; ++r) {
        const int i0 = r * 16;
        v8f acc = {};
#pragma unroll
        for (int kk = 0; kk < 16; ++kk) {
            const int kb = kk * 4 + half * 2;
            v2f a;
            a.x = D[(size_t)kb       * NDICT + i0 + lid];
            a.y = D[(size_t)(kb + 1) * NDICT + i0 + lid];
            acc = __builtin_amdgcn_wmma_f32_16x16x4_f32(
                false, a, false, bx[kk], (short)0, acc, false, false);
        }
#pragma unroll
        for (int v = 0; v < 8; ++v) {
            const int m = v + 8 * half;   // row i within tile
            sDtx[lid * 513 + i0 + m] = acc[v];  // lid = column within block
        }
    }
    __syncthreads();  // single-wave WG: S_NOP; LDS in-order per wave anyway

    // ---- Phase B: OMP iterations, one column at a time, lanes own rows ----
    for (int c = 0; c < 16; ++c) {
        float dtxr[16];
#pragma unroll
        for (int t = 0; t < 16; ++t)
            dtxr[t] = sDtx[c * 513 + lane + 32 * t];

        int   active[SPAR];
        float sol[SPAR];
        float Ga[SPAR][SPAR];
        float rhs[SPAR];

#pragma unroll
        for (int k = 0; k < SPAR; ++k) {
            // corr_i = dtx_i - sum_{j<k} sol_j * G[active_j][i]; masked abs-argmax
            float bestv = -1.0f;
            int   besti = 0;
#pragma unroll
            for (int t = 0; t < 16; ++t) {
                const int i = lane + 32 * t;
                float val = dtxr[t];
#pragma unroll
                for (int j = 0; j < SPAR; ++j)
                    if (j < k)
                        val = fmaf(-sol[j], G[(size_t)active[j] * NDICT + i], val);
                bool masked = false;
#pragma unroll
                for (int j = 0; j < SPAR; ++j)
                    if (j < k) masked |= (i == active[j]);
                const float av = masked ? 0.0f : fabsf(val);
                if (av > bestv) { bestv = av; besti = i; }  // keeps smallest i on ties
            }
            // wave32 argmax reduction (tie -> smaller index, matches jnp.argmax)
#pragma unroll
            for (int off = 16; off > 0; off >>= 1) {
                const float ov = __shfl_xor(bestv, off, 32);
                const int   oi = __shfl_xor(besti, off, 32);
                if (ov > bestv || (ov == bestv && oi < besti)) { bestv = ov; besti = oi; }
            }
            const int idx = besti;
            active[k] = idx;

            // extend Gram subsystem (scalar broadcast loads; G symmetric)
#pragma unroll
            for (int j = 0; j < SPAR; ++j)
                if (j <= k) {
                    const float g = G[(size_t)idx * NDICT + active[j]];
                    Ga[k][j] = g;
                    Ga[j][k] = g;
                }
            Ga[k][k] += DIAG_EPS;
            rhs[k] = sDtx[c * 513 + idx];

            // solve (k+1)x(k+1) SPD system: unpivoted Gaussian elimination,
            // fully unrolled (all indices compile-time -> stays in VGPRs)
            float A[SPAR][SPAR + 1];
#pragma unroll
            for (int r2 = 0; r2 < SPAR; ++r2) {
#pragma unroll
                for (int c2 = 0; c2 < SPAR; ++c2)
                    A[r2][c2] = (r2 <= k && c2 <= k) ? Ga[r2][c2] : (r2 == c2 ? 1.0f : 0.0f);
                A[r2][SPAR] = (r2 <= k) ? rhs[r2] : 0.0f;
            }
#pragma unroll
            for (int p = 0; p < SPAR; ++p) {
                const float inv = 1.0f / A[p][p];
#pragma unroll
                for (int r2 = p + 1; r2 < SPAR; ++r2) {
                    const float f = A[r2][p] * inv;
#pragma unroll
                    for (int c2 = p; c2 <= SPAR; ++c2)
                        A[r2][c2] = fmaf(-f, A[p][c2], A[r2][c2]);
                }
            }
#pragma unroll
            for (int r2 = SPAR - 1; r2 >= 0; --r2) {
                float s = A[r2][SPAR];
#pragma unroll
                for (int c2 = r2 + 1; c2 < SPAR; ++c2)
                    s = fmaf(-A[r2][c2], sol[c2], s);
                sol[r2] = s / A[r2][r2];
            }
        }

        // final scatter: 5 nonzeros into coeff (N, B)
        if (lane == 0) {
#pragma unroll
            for (int j = 0; j < SPAR; ++j)
                out[(size_t)active[j] * BDIM + (col0 + c)] = sol[j];
        }
    }
}

// ---------------------------------------------------------------------------
extern "C" void kernel_launch(void* const* d_in, const int* in_sizes, int n_in,
                              void* d_out, int out_size, void* d_ws, size_t ws_size,
                              hipStream_t stream) {
    const float* X = (const float*)d_in[0];   // (64, 65536)
    const float* D = (const float*)d_in[1];   // (64, 512)
    float* out = (float*)d_out;               // (512, 65536)
    float* G   = (float*)d_ws;                // 512*512 floats = 1 MB scratch

    // 1) zero the output (128 MB)
    const int n4 = out_size / 4;
    zero_out_kernel<<<4096, 256, 0, stream>>>((float4*)out, n4);

    // 2) gram = D^T D (512x512), fp32 WMMA
    gram_kernel<<<dim3(32, 32), 32, 0, stream>>>(D, G);

    // 3) fused dtx + batch OMP: one wave per 16 columns
    omp_kernel<<<BDIM / 16, 32, 0, stream>>>(X, D, G, out);
}